// TGN_67104569033114
// MI455X (gfx1250) — compile-verified
//
#include <hip/hip_runtime.h>

typedef float v2f __attribute__((ext_vector_type(2)));
typedef float v8f __attribute__((ext_vector_type(8)));

// D = A(16x4 f32) * B(4x16 f32) + C(16x16 f32)  -> v_wmma_f32_16x16x4_f32
static __device__ __forceinline__ v8f wmma4(v2f a, v2f b, v8f c) {
  return __builtin_amdgcn_wmma_f32_16x16x4_f32(
      /*neg_a=*/false, a, /*neg_b=*/false, b,
      /*c_mod=*/(short)0, c, /*reuse_a=*/false, /*reuse_b=*/false);
}

// Guaranteed hardware fp32 atomic (no-return -> STOREcnt; s_endpgm waits idle).
static __device__ __forceinline__ void global_fadd(float* p, float v) {
  unsigned long long a = (unsigned long long)p;
  asm volatile("global_atomic_add_f32 %0, %1, off" : : "v"(a), "v"(v) : "memory");
}

// ---------------------------------------------------------------------------
// Kernel A: h[B,32] = mem[ids][:,32] @ W[32,32]
// Persistent waves, W fragments hoisted to registers (32 VGPRs).
// mem rows + ids are single-use: non-temporal so they don't evict the
// L2-resident edge-feature table (128 MB of 192 MB L2).
// ---------------------------------------------------------------------------
__global__ void __launch_bounds__(128)
node_transform_kernel(const float* __restrict__ mem, const int* __restrict__ ids,
                      const float* __restrict__ W, float* __restrict__ h, int ntiles) {
  const int lane = threadIdx.x & 31;
  const int m  = lane & 15;
  const int hi = lane >> 4;
  const int wid = blockIdx.x * 4 + (threadIdx.x >> 5);
  const int nw  = gridDim.x * 4;

  v2f Wb0[8], Wb1[8];                           // loop-invariant B fragments
#pragma unroll
  for (int k = 0; k < 8; ++k) {
    const int ka = k * 4 + hi * 2;
    Wb0[k].x = W[(ka + 0) * 32 + m];      Wb0[k].y = W[(ka + 1) * 32 + m];
    Wb1[k].x = W[(ka + 0) * 32 + 16 + m]; Wb1[k].y = W[(ka + 1) * 32 + 16 + m];
  }

  for (int tile = wid; tile < ntiles; tile += nw) {
    const long long node = __builtin_nontemporal_load(ids + tile * 16 + m);
    const float* mrow = mem + node * 32;
    v8f c0 = {}, c1 = {};
#pragma unroll
    for (int k = 0; k < 8; ++k) {
      v2f a = __builtin_nontemporal_load((const v2f*)(mrow + k * 4 + hi * 2));
      c0 = wmma4(a, Wb0[k], c0);
      c1 = wmma4(a, Wb1[k], c1);
    }
    float* hb = h + (size_t)tile * 16 * 32;     // RT store: keep h in L2 for gathers
#pragma unroll
    for (int v = 0; v < 8; ++v) {
      const int r = v + hi * 8;
      hb[r * 32 + m]      = c0[v];
      hb[r * 32 + 16 + m] = c1[v];
    }
  }
}

// ---------------------------------------------------------------------------
// Kernel B: msg = h_src[src] + feat[idx]@We + be + cos(t*w+b) -> atomic agg[dst]
// Persistent waves; We fragments hoisted (64 VGPRs); feature rows default-
// temporal (4x reuse, L2-resident); index/time streams non-temporal;
// next tile's gathered rows prefetched (global_prefetch_b8) under the WMMAs.
// ---------------------------------------------------------------------------
__global__ void __launch_bounds__(128)
edge_message_kernel(const float* __restrict__ hsrc, const float* __restrict__ efeat,
                    const int* __restrict__ esrc, const int* __restrict__ edst,
                    const int* __restrict__ eidx, const float* __restrict__ etime,
                    const float* __restrict__ We, const float* __restrict__ be,
                    const float* __restrict__ tw, const float* __restrict__ tb,
                    float* __restrict__ agg, int ntiles) {
  const int lane = threadIdx.x & 31;
  const int m  = lane & 15;
  const int hi = lane >> 4;
  const int wid = blockIdx.x * 4 + (threadIdx.x >> 5);
  const int nw  = gridDim.x * 4;

  v2f Wb0[16], Wb1[16];                         // loop-invariant B fragments
#pragma unroll
  for (int k = 0; k < 16; ++k) {
    const int ka = k * 4 + hi * 2;
    Wb0[k].x = We[(ka + 0) * 32 + m];      Wb0[k].y = We[(ka + 1) * 32 + m];
    Wb1[k].x = We[(ka + 0) * 32 + 16 + m]; Wb1[k].y = We[(ka + 1) * 32 + 16 + m];
  }
  const float be0 = be[m],      be1 = be[16 + m];
  const float tw0 = tw[m],      tw1 = tw[16 + m];
  const float tb0 = tb[m],      tb1 = tb[16 + m];

  if (wid >= ntiles) return;
  long long fid = __builtin_nontemporal_load(eidx + wid * 16 + m);
  for (int tile = wid; tile < ntiles; tile += nw) {
    const int tnext = tile + nw;
    long long fid_next = fid;
    if (tnext < ntiles) {                       // prefetch next gathered row (256B)
      fid_next = __builtin_nontemporal_load(eidx + tnext * 16 + m);
      __builtin_prefetch(efeat + fid_next * 64, 0, 3);
      __builtin_prefetch(efeat + fid_next * 64 + 32, 0, 3);
    }

    const float* frow = efeat + fid * 64;
    v8f c0 = {}, c1 = {};
#pragma unroll
    for (int k = 0; k < 16; ++k) {              // K = 64 -> 32 WMMAs / 16 edges
      v2f a = *(const v2f*)(frow + k * 4 + hi * 2);
      c0 = wmma4(a, Wb0[k], c0);
      c1 = wmma4(a, Wb1[k], c1);
    }

#pragma unroll
    for (int v = 0; v < 8; ++v) {
      const int e = tile * 16 + v + hi * 8;
      const float t    = __builtin_nontemporal_load(etime + e);
      const long long s = __builtin_nontemporal_load(esrc + e);
      const long long d = __builtin_nontemporal_load(edst + e);
      const float m0 = c0[v] + be0 + __cosf(t * tw0 + tb0) + hsrc[s * 32 + m];
      const float m1 = c1[v] + be1 + __cosf(t * tw1 + tb1) + hsrc[s * 32 + 16 + m];
      global_fadd(agg + d * 32 + m,      m0);   // L2-resident segment_sum
      global_fadd(agg + d * 32 + 16 + m, m1);
    }
    fid = fid_next;
  }
}

// ---------------------------------------------------------------------------
// Kernel C: z=[relu(hu+aggu)|relu(hi+aggi)], out = relu(relu(z@W1+b1)@W2+b2)@W3+b3
// Block-uniform grid-stride (clamped tile keeps __syncthreads convergent).
// h/agg are dead after this read -> non-temporal.
// ---------------------------------------------------------------------------
__global__ void __launch_bounds__(128)
decoder_kernel(const float* __restrict__ hu, const float* __restrict__ au,
               const float* __restrict__ hiT, const float* __restrict__ ai,
               const float* __restrict__ W1, const float* __restrict__ b1,
               const float* __restrict__ W2, const float* __restrict__ b2,
               const float* __restrict__ W3, const float* __restrict__ b3,
               float* __restrict__ out, int ntiles) {
  __shared__ float lds_x1[4][16 * 64];
  __shared__ float lds_x2[4][16 * 16];
  const int lane = threadIdx.x & 31;
  const int wv   = threadIdx.x >> 5;
  const int m  = lane & 15;
  const int hi = lane >> 4;

  v2f W2b[16];                                   // hoisted W2 fragments (32 VGPRs)
#pragma unroll
  for (int k = 0; k < 16; ++k) {
    const int ka = k * 4 + hi * 2;
    W2b[k].x = W2[(ka + 0) * 16 + m];
    W2b[k].y = W2[(ka + 1) * 16 + m];
  }
  const float b2m = b2[m];

  for (int base = blockIdx.x * 4; base < ntiles; base += gridDim.x * 4) {
    int tile = base + wv;
    if (tile >= ntiles) tile = ntiles - 1;       // clamp: barriers stay uniform
    const int r = tile * 16 + m;

    // ---- x1 = relu(z @ W1 + b1), K=64, N=64 (4 tiles) ----
    v8f d0 = {}, d1 = {}, d2 = {}, d3 = {};
#pragma unroll
    for (int k = 0; k < 16; ++k) {
      const int ka = k * 4 + hi * 2;
      v2f a;
      if (ka < 32) {
        a.x = fmaxf(__builtin_nontemporal_load(hu + (size_t)r * 32 + ka) +
                    __builtin_nontemporal_load(au + (size_t)r * 32 + ka), 0.f);
        a.y = fmaxf(__builtin_nontemporal_load(hu + (size_t)r * 32 + ka + 1) +
                    __builtin_nontemporal_load(au + (size_t)r * 32 + ka + 1), 0.f);
      } else {
        const int kb = ka - 32;
        a.x = fmaxf(__builtin_nontemporal_load(hiT + (size_t)r * 32 + kb) +
                    __builtin_nontemporal_load(ai + (size_t)r * 32 + kb), 0.f);
        a.y = fmaxf(__builtin_nontemporal_load(hiT + (size_t)r * 32 + kb + 1) +
                    __builtin_nontemporal_load(ai + (size_t)r * 32 + kb + 1), 0.f);
      }
      v2f b;
      b.x = W1[(ka + 0) * 64 + m];      b.y = W1[(ka + 1) * 64 + m];      d0 = wmma4(a, b, d0);
      b.x = W1[(ka + 0) * 64 + 16 + m]; b.y = W1[(ka + 1) * 64 + 16 + m]; d1 = wmma4(a, b, d1);
      b.x = W1[(ka + 0) * 64 + 32 + m]; b.y = W1[(ka + 1) * 64 + 32 + m]; d2 = wmma4(a, b, d2);
      b.x = W1[(ka + 0) * 64 + 48 + m]; b.y = W1[(ka + 1) * 64 + 48 + m]; d3 = wmma4(a, b, d3);
    }
#pragma unroll
    for (int v = 0; v < 8; ++v) {                // C layout -> LDS -> A layout
      const int rl = v + hi * 8;
      lds_x1[wv][rl * 64 +  0 + m] = fmaxf(d0[v] + b1[ 0 + m], 0.f);
      lds_x1[wv][rl * 64 + 16 + m] = fmaxf(d1[v] + b1[16 + m], 0.f);
      lds_x1[wv][rl * 64 + 32 + m] = fmaxf(d2[v] + b1[32 + m], 0.f);
      lds_x1[wv][rl * 64 + 48 + m] = fmaxf(d3[v] + b1[48 + m], 0.f);
    }
    __syncthreads();

    // ---- x2 = relu(x1 @ W2 + b2), K=64, N=16 ----
    v8f e = {};
#pragma unroll
    for (int k = 0; k < 16; ++k) {
      const int ka = k * 4 + hi * 2;
      v2f a;
      a.x = lds_x1[wv][m * 64 + ka];
      a.y = lds_x1[wv][m * 64 + ka + 1];
      e = wmma4(a, W2b[k], e);
    }
#pragma unroll
    for (int v = 0; v < 8; ++v)
      lds_x2[wv][(v + hi * 8) * 16 + m] = fmaxf(e[v] + b2m, 0.f);
    __syncthreads();

    // ---- out = x2 @ W3 + b3 : 16x1, one lane per row ----
    if (lane < 16) {
      float acc = b3[0];
#pragma unroll
      for (int n = 0; n < 16; ++n) acc += lds_x2[wv][lane * 16 + n] * W3[n];
      __builtin_nontemporal_store(acc, out + tile * 16 + lane);
    }
  }
}

// ---------------------------------------------------------------------------
extern "C" void kernel_launch(void* const* d_in, const int* in_sizes, int n_in,
                              void* d_out, int out_size, void* d_ws, size_t ws_size,
                              hipStream_t stream) {
  const float* mem_user = (const float*)d_in[0];
  const float* mem_item = (const float*)d_in[1];
  const int* node_ids_user = (const int*)d_in[2];
  const int* node_ids_item = (const int*)d_in[3];
  const int* edge_src_ui = (const int*)d_in[4];
  const int* edge_dst_ui = (const int*)d_in[5];
  const int* edge_src_iu = (const int*)d_in[6];
  const int* edge_dst_iu = (const int*)d_in[7];
  const int* edge_idx_ui = (const int*)d_in[8];
  const int* edge_idx_iu = (const int*)d_in[9];
  const float* rt_ui = (const float*)d_in[10];
  const float* rt_iu = (const float*)d_in[11];
  const float* efeat = (const float*)d_in[12];
  const float* W_user = (const float*)d_in[13];
  const float* W_item = (const float*)d_in[14];
  const float* We_ui = (const float*)d_in[15];
  const float* be_ui = (const float*)d_in[16];
  const float* We_iu = (const float*)d_in[17];
  const float* be_iu = (const float*)d_in[18];
  const float* time_w = (const float*)d_in[19];
  const float* time_b = (const float*)d_in[20];
  const float* dW1 = (const float*)d_in[21];
  const float* db1 = (const float*)d_in[22];
  const float* dW2 = (const float*)d_in[23];
  const float* db2 = (const float*)d_in[24];
  const float* dW3 = (const float*)d_in[25];
  const float* db3 = (const float*)d_in[26];

  const int B = in_sizes[2];   // 262144
  const int E = in_sizes[4];   // 1048576

  // workspace: h_user | h_item | agg_user | agg_item, each B*32 f32 (134 MB)
  float* h_user   = (float*)d_ws;
  float* h_item   = h_user + (size_t)B * 32;
  float* agg_user = h_item + (size_t)B * 32;
  float* agg_item = agg_user + (size_t)B * 32;

  hipMemsetAsync(agg_user, 0, (size_t)B * 32 * 2 * sizeof(float), stream);

  const int ntilesB = B / 16;
  const int ntilesE = E / 16;
  const int blocksA = 1024;    // 4096 persistent waves, grid-stride
  const int blocksE = 4096;    // 16384 persistent waves
  const int blocksC = 1024;

  node_transform_kernel<<<blocksA, 128, 0, stream>>>(mem_user, node_ids_user, W_user, h_user, ntilesB);
  node_transform_kernel<<<blocksA, 128, 0, stream>>>(mem_item, node_ids_item, W_item, h_item, ntilesB);

  // user->item edges aggregate onto item nodes; item->user onto user nodes
  edge_message_kernel<<<blocksE, 128, 0, stream>>>(h_user, efeat, edge_src_ui, edge_dst_ui,
      edge_idx_ui, rt_ui, We_ui, be_ui, time_w, time_b, agg_item, ntilesE);
  edge_message_kernel<<<blocksE, 128, 0, stream>>>(h_item, efeat, edge_src_iu, edge_dst_iu,
      edge_idx_iu, rt_iu, We_iu, be_iu, time_w, time_b, agg_user, ntilesE);

  decoder_kernel<<<blocksC, 128, 0, stream>>>(h_user, agg_user, h_item, agg_item,
      dW1, db1, dW2, db2, dW3, db3, (float*)d_out, ntilesB);
}